// MorphLoss_2224793060017
// MI455X (gfx1250) — compile-verified
//
#include <hip/hip_runtime.h>
#include <hip/hip_bf16.h>

// MorphLoss on MI455X (gfx1250, wave32).
//   K1: binarize + vertical two-scan distance -> g2 = g*g   [4][512][512]
//   K2: column lower-envelope (min-plus VALU core), g2 tile staged in LDS via
//       gfx1250 async direct-to-LDS copies (ASYNCcnt), 4-way min ILP -> d
//   K3: masked sums via V_WMMA_F32_16X16X4_F32 (ones-matrix GEMM reduction),
//       branchless selectors, b64 vector loads
//   K4: combine scalars into the loss.

typedef float v2f __attribute__((ext_vector_type(2)));
typedef float v8f __attribute__((ext_vector_type(8)));

#define HH 512
#define WW 512
#define NPIX (HH * WW)
#define NB 32            // reduction blocks per phase
#define INF_D 1024.0f    // reference INF = H + W

__global__ void edt_vertical_kernel(const float* __restrict__ img,
                                    const float* __restrict__ tgt,
                                    float* __restrict__ g2) {
    int t = blockIdx.x * blockDim.x + threadIdx.x;   // 0..2047
    int p = t >> 9;                                  // phase 0..3
    int j = t & (WW - 1);                            // column
    const float* src = (p & 1) ? tgt : img;
    bool sep = (p >= 2);
    float* out = g2 + p * NPIX;

    // forward scan (top -> bottom)
    float carry = INF_D;
    for (int k = 0; k < HH; ++k) {
        bool ph = (src[k * WW + j] > 0.22f) != sep;
        float v = ph ? (carry + 1.0f) : 0.0f;
        out[k * WW + j] = v;
        carry = v;
    }
    // backward scan, combine, clamp to INF, square
    carry = INF_D;
    for (int k = HH - 1; k >= 0; --k) {
        bool ph = (src[k * WW + j] > 0.22f) != sep;
        float v = ph ? (carry + 1.0f) : 0.0f;
        carry = v;
        float g = fminf(fminf(out[k * WW + j], v), INF_D);
        out[k * WW + j] = g * g;
    }
}

// Lower envelope: d2[i,j] = min_k g2[k,j] + (i-k)^2 ; d = sqrt(d2)
// Block: 256 threads = 16 cols x 16 rows-of-i, loops over 32 i-tiles.
// 512x16 f32 tile (32KB) pulled into LDS with async direct-to-LDS b128 copies.
__global__ void edt_envelope_kernel(const float* __restrict__ g2,
                                    float* __restrict__ dist) {
    __shared__ __align__(16) float g2s[HH][16];
    int p  = blockIdx.y;             // phase
    int jb = blockIdx.x;             // column block 0..31
    int tj = threadIdx.x & 15;
    int ti = threadIdx.x >> 4;

    const float* gbase = g2 + p * NPIX + jb * 16;
    // LDS byte offset of the tile: flat shared addresses carry the wave-relative
    // LDS offset in their low 32 bits (aperture rules, CDNA5 ISA).
    uint32_t lds_base = (uint32_t)(uintptr_t)(&g2s[0][0]);
    #pragma unroll
    for (int v = 0; v < 8; ++v) {
        int e = ((int)threadIdx.x + v * 256) * 4;   // float index in tile
        int k = e >> 4;
        int c = e & 15;
        uint64_t ga = (uint64_t)(uintptr_t)(gbase + k * WW + c);
        uint32_t la = lds_base + (uint32_t)e * 4u;
        asm volatile("global_load_async_to_lds_b128 %0, %1, off"
                     :: "v"(la), "v"(ga) : "memory");
    }
    asm volatile("s_wait_asynccnt 0x0" ::: "memory");
    __syncthreads();

    float* dp = dist + p * NPIX + jb * 16;
    for (int itile = 0; itile < HH / 16; ++itile) {
        int i = itile * 16 + ti;
        float dlt = (float)i;            // i - k for the current group of 8
        // 4 independent min chains for VALU ILP (min is exact & associative here)
        float b0 = 3.4e38f, b1 = 3.4e38f, b2 = 3.4e38f, b3 = 3.4e38f;
        for (int k = 0; k < HH; k += 8) {
            float d0 = dlt - 0.0f, d1 = dlt - 1.0f, d2 = dlt - 2.0f, d3 = dlt - 3.0f;
            float d4 = dlt - 4.0f, d5 = dlt - 5.0f, d6 = dlt - 6.0f, d7 = dlt - 7.0f;
            b0 = fminf(b0, fmaf(d0, d0, g2s[k + 0][tj]));
            b1 = fminf(b1, fmaf(d1, d1, g2s[k + 1][tj]));
            b2 = fminf(b2, fmaf(d2, d2, g2s[k + 2][tj]));
            b3 = fminf(b3, fmaf(d3, d3, g2s[k + 3][tj]));
            b0 = fminf(b0, fmaf(d4, d4, g2s[k + 4][tj]));
            b1 = fminf(b1, fmaf(d5, d5, g2s[k + 5][tj]));
            b2 = fminf(b2, fmaf(d6, d6, g2s[k + 6][tj]));
            b3 = fminf(b3, fmaf(d7, d7, g2s[k + 7][tj]));
            dlt -= 8.0f;
        }
        float best = fminf(fminf(b0, b1), fminf(b2, b3));
        dp[i * WW + tj] = sqrtf(best);
    }
}

// Masked sums via WMMA: C += ones(16x4) x B(4x16). With all-ones A,
// sum(all C entries) = 16 * sum(all B entries) regardless of B lane layout,
// so each lane packs 2 arbitrary data values per op. f32 A/B/C: exact-width sums.
__global__ void masked_sum_wmma_kernel(const float* __restrict__ dist,
                                       const float* __restrict__ img,
                                       const float* __restrict__ tgt,
                                       const float* __restrict__ mask,
                                       float* __restrict__ partials) {
    int p   = blockIdx.y;
    int blk = blockIdx.x;                       // 0..NB-1
    const float* src = (p & 1) ? tgt : img;
    bool sep = (p >= 2);
    const float* dp = dist + p * NPIX;

    int wave = threadIdx.x >> 5;                // 0..7
    int lane = threadIdx.x & 31;
    int base = blk * (NPIX / NB) + wave * 1024; // 1024 elems per wave

    v2f ones2; ones2[0] = 1.0f; ones2[1] = 1.0f;
    v8f accD = {};   // sum of d * sel
    v8f accS = {};   // sum of sel

    for (int c = 0; c < 16; ++c) {              // 16 chunks of 64 elems
        int idx = base + c * 64 + lane * 2;     // even -> 8B-aligned v2f loads
        v2f dv = *(const v2f*)(dp + idx);
        v2f sv = *(const v2f*)(src + idx);
        v2f mv = *(const v2f*)(mask + idx);
        // branchless selector: compare & compare (no short-circuit, no EXEC games)
        float s0 = (float)(int)(((sv[0] > 0.22f) != sep) & (mv[0] > 0.5f));
        float s1 = (float)(int)(((sv[1] > 0.22f) != sep) & (mv[1] > 0.5f));
        v2f bD; bD[0] = dv[0] * s0; bD[1] = dv[1] * s1;
        v2f bS; bS[0] = s0;         bS[1] = s1;
        accD = __builtin_amdgcn_wmma_f32_16x16x4_f32(
            false, ones2, false, bD, (short)0, accD, false, false);
        accS = __builtin_amdgcn_wmma_f32_16x16x4_f32(
            false, ones2, false, bS, (short)0, accS, false, false);
    }

    float sd = 0.0f, ss = 0.0f;
    #pragma unroll
    for (int r = 0; r < 8; ++r) { sd += accD[r]; ss += accS[r]; }
    // wave32 butterfly reduce -> every lane holds the full 256-entry C sum
    for (int off = 16; off > 0; off >>= 1) {
        sd += __shfl_xor(sd, off, 32);
        ss += __shfl_xor(ss, off, 32);
    }

    __shared__ float wsd[8], wss[8];
    if (lane == 0) { wsd[wave] = sd; wss[wave] = ss; }
    __syncthreads();
    if (threadIdx.x == 0) {
        float td = 0.0f, ts = 0.0f;
        for (int wv = 0; wv < 8; ++wv) { td += wsd[wv]; ts += wss[wv]; }
        partials[(p * NB + blk) * 2 + 0] = td * (1.0f / 16.0f);
        partials[(p * NB + blk) * 2 + 1] = ts * (1.0f / 16.0f);
    }
}

__global__ void final_combine_kernel(const float* __restrict__ partials,
                                     float* __restrict__ out) {
    if (threadIdx.x != 0 || blockIdx.x != 0) return;
    float th[4];
    for (int p = 0; p < 4; ++p) {
        float sd = 0.0f, ss = 0.0f;
        for (int b = 0; b < NB; ++b) {
            sd += partials[(p * NB + b) * 2 + 0];
            ss += partials[(p * NB + b) * 2 + 1];
        }
        th[p] = 2.0f * 10.5f * sd / fmaxf(ss, 1.0f);
    }
    // ((a-mu)/sd - (b-mu)/sd)^2 == ((a-b)/sd)^2
    float l1 = (th[0] - th[1]) / 5.2874f;    // thickness (fg phases)
    float l7 = (th[2] - th[3]) / 46.1809f;   // separation (bg phases)
    out[0] = 0.5f * (l1 * l1 + l7 * l7);
}

extern "C" void kernel_launch(void* const* d_in, const int* in_sizes, int n_in,
                              void* d_out, int out_size, void* d_ws, size_t ws_size,
                              hipStream_t stream) {
    const float* img  = (const float*)d_in[0];
    const float* mask = (const float*)d_in[1];
    const float* tgt  = (const float*)d_in[2];

    float* g2       = (float*)d_ws;            // 4 * NPIX floats (4 MB)
    float* dist     = g2 + 4 * NPIX;           // 4 * NPIX floats (4 MB)
    float* partials = dist + 4 * NPIX;         // 4 * NB * 2 floats

    edt_vertical_kernel<<<8, 256, 0, stream>>>(img, tgt, g2);

    dim3 gEnv(WW / 16, 4);
    edt_envelope_kernel<<<gEnv, 256, 0, stream>>>(g2, dist);

    dim3 gRed(NB, 4);
    masked_sum_wmma_kernel<<<gRed, 256, 0, stream>>>(dist, img, tgt, mask, partials);

    final_combine_kernel<<<1, 64, 0, stream>>>(partials, (float*)d_out);
}